// Latent_56693568307582
// MI455X (gfx1250) — compile-verified
//
#include <hip/hip_runtime.h>
#include <math.h>

// MI455X / gfx1250, wave32. f32 WMMA path: V_WMMA_F32_16X16X4_F32.

typedef float v2f __attribute__((ext_vector_type(2)));
typedef float v4f __attribute__((ext_vector_type(4)));
typedef float v8f __attribute__((ext_vector_type(8)));

// Problem constants: N=16, z_dim=64, H=W=32 -> P=16384; L=4, M=1024, D=16.
#define LCNT   4
#define MCNT   1024
#define DCNT   16
#define PCNT   16384
#define HWCNT  1024      // H*W
#define NCHUNK 16        // one chunk per n (batch index)
#define MTILES 64        // 1024 / 16
#define PTILES 64        // p-tiles per chunk
#define TILE_F 320       // packed floats per tile: 256 A + 64 A2
#define LOG2E  1.44269504088896340736f

// ---------------------------------------------------------------------------
// Prep 1: eS = (-2*a')*e, esqS = a'*||e||^2, zsqS = a'*||z||^2,
// with a' = alpha*log2(e) so the WMMA accumulator is already in log2 domain
// and the epilogue is a bare v_exp_f32 (exp2).
// ---------------------------------------------------------------------------
__global__ __launch_bounds__(256) void prep_kernel(
    const float* __restrict__ z, const float* __restrict__ e,
    const float* __restrict__ log_sigma,
    float* __restrict__ esqS, float* __restrict__ eS, float* __restrict__ zsqS)
{
  int idx = blockIdx.x * 256 + threadIdx.x;          // 0 .. 65535
  float ls = log_sigma[0];
  float a2 = -0.5f * __expf(-2.0f * ls) * LOG2E;     // alpha * log2(e)

  int l  = idx >> 14;
  int p  = idx & (PCNT - 1);
  int n  = p >> 10;
  int hw = p & (HWCNT - 1);
  const float* zp = z + n * 65536 + l * 16384 + hw;  // + d*1024
  float s = 0.0f;
#pragma unroll
  for (int d = 0; d < DCNT; ++d) { float v = zp[d * 1024]; s += v * v; }
  zsqS[idx] = a2 * s;

  if (idx < LCNT * MCNT) {
    const float* ep = e + idx * DCNT;
    float t = 0.0f;
#pragma unroll
    for (int d = 0; d < DCNT; ++d) {
      float v = ep[d];
      eS[idx * DCNT + d] = (-2.0f * a2) * v;
      t += v * v;
    }
    esqS[idx] = a2 * t;
  }
}

// ---------------------------------------------------------------------------
// Prep 2: pack z into per-lane WMMA A layout.
// Tile (l, chunk, t): 320 floats = [lane][8 A values] (256) + [lane][2 A2] (64).
// Lane (lo=lane&15, hi=lane>>4) owns A row lo, d = 2*hi + {0,1,4,5,8,9,12,13}.
// A2: lanes 0-15 -> (a'*zsq_p, 1), lanes 16-31 -> (0, 0).
// One thread per (l, chunk, t, lane): 131072 threads.
// ---------------------------------------------------------------------------
__global__ __launch_bounds__(256) void pack_kernel(
    const float* __restrict__ z, const float* __restrict__ zsqS,
    float* __restrict__ packedA)
{
  int tid   = blockIdx.x * 256 + threadIdx.x;        // 0 .. 131071
  int l     = tid >> 15;
  int chunk = (tid >> 11) & (NCHUNK - 1);
  int t     = (tid >> 5) & (PTILES - 1);
  int lane  = tid & 31;
  int lo    = lane & 15;
  int hi    = lane >> 4;

  const float* src = z + chunk * 65536 + l * 16384 + hi * 2048 + 16 * t + lo;
  float* dst = packedA + (((l * NCHUNK + chunk) * PTILES + t) * TILE_F);

  v4f x, y;
  x[0] = src[0];          x[1] = src[1024];
  x[2] = src[4 * 1024];   x[3] = src[5 * 1024];
  y[0] = src[8 * 1024];   y[1] = src[9 * 1024];
  y[2] = src[12 * 1024];  y[3] = src[13 * 1024];
  *(v4f*)(dst + lane * 8)     = x;
  *(v4f*)(dst + lane * 8 + 4) = y;

  float zq = zsqS[l * PCNT + chunk * HWCNT + 16 * t + lo];   // already a'-scaled
  v2f aS;
  aS.x = hi ? 0.0f : zq;
  aS.y = hi ? 0.0f : 1.0f;
  *(v2f*)(dst + 256 + lane * 2) = aS;
}

// ---------------------------------------------------------------------------
// Main (packed): wave owns (l, m-tile, chunk). Per p-tile:
// 2x b128 + 1x b64 loads, 5 chained f32 WMMAs, 8 exp2, 8 adds.
// Accumulator holds log2(e)*alpha*d2, so exp2 gives exp(alpha*d2).
// ---------------------------------------------------------------------------
__global__ __launch_bounds__(256) void vq_main_packed(
    const float* __restrict__ packedA, const float* __restrict__ eS,
    const float* __restrict__ esqS, float* __restrict__ Spart)
{
  int tid  = blockIdx.x * 256 + threadIdx.x;
  int wid  = tid >> 5;
  int lane = threadIdx.x & 31;
  int lo   = lane & 15;
  int hi   = lane >> 4;

  int chunk = wid & (NCHUNK - 1);
  int mt    = (wid >> 4) & (MTILES - 1);
  int l     = wid >> 10;
  int m0    = mt * 16;

  const float* pt = packedA + (size_t)(l * NCHUNK + chunk) * (PTILES * TILE_F);
  const float* pa = pt + lane * 8;
  const float* ps = pt + 256 + lane * 2;

  const float* eb = eS + (l * MCNT + m0 + lo) * DCNT + 2 * hi;
  v2f b0 = *(const v2f*)(eb + 0);
  v2f b1 = *(const v2f*)(eb + 4);
  v2f b2 = *(const v2f*)(eb + 8);
  v2f b3 = *(const v2f*)(eb + 12);

  float esq = esqS[l * MCNT + m0 + lo];
  v2f bS;
  bS.x = hi ? 0.0f : 1.0f;
  bS.y = hi ? 0.0f : esq;

  float p0 = 0.0f, p1 = 0.0f;
#pragma unroll 2
  for (int t = 0; t < PTILES; ++t) {
    v4f x  = *(const v4f*)(pa + t * TILE_F);
    v4f y  = *(const v4f*)(pa + t * TILE_F + 4);
    v2f aS = *(const v2f*)(ps + t * TILE_F);

    v2f a0; a0.x = x[0]; a0.y = x[1];
    v2f a1; a1.x = x[2]; a1.y = x[3];
    v2f a2; a2.x = y[0]; a2.y = y[1];
    v2f a3; a3.x = y[2]; a3.y = y[3];

    v8f c = {0.f, 0.f, 0.f, 0.f, 0.f, 0.f, 0.f, 0.f};
    c = __builtin_amdgcn_wmma_f32_16x16x4_f32(false, a0, false, b0, (short)0, c, false, false);
    c = __builtin_amdgcn_wmma_f32_16x16x4_f32(false, a1, false, b1, (short)0, c, false, false);
    c = __builtin_amdgcn_wmma_f32_16x16x4_f32(false, a2, false, b2, (short)0, c, false, false);
    c = __builtin_amdgcn_wmma_f32_16x16x4_f32(false, a3, false, b3, (short)0, c, false, false);
    c = __builtin_amdgcn_wmma_f32_16x16x4_f32(false, aS, false, bS, (short)0, c, false, false);

    p0 += __builtin_amdgcn_exp2f(c[0]) + __builtin_amdgcn_exp2f(c[2])
        + __builtin_amdgcn_exp2f(c[4]) + __builtin_amdgcn_exp2f(c[6]);
    p1 += __builtin_amdgcn_exp2f(c[1]) + __builtin_amdgcn_exp2f(c[3])
        + __builtin_amdgcn_exp2f(c[5]) + __builtin_amdgcn_exp2f(c[7]);
  }
  float partial = p0 + p1;
  partial += __shfl_xor(partial, 16, 32);
  if (lane < 16)
    Spart[chunk * (LCNT * MCNT) + l * MCNT + m0 + lane] = partial;
}

// ---------------------------------------------------------------------------
// Fallback main (no packed buffer; ~0.8 MB workspace total). Same math.
// ---------------------------------------------------------------------------
__global__ __launch_bounds__(256) void vq_main_fallback(
    const float* __restrict__ z,    const float* __restrict__ eS,
    const float* __restrict__ esqS, const float* __restrict__ zsqS,
    float* __restrict__ Spart)
{
  int tid  = blockIdx.x * 256 + threadIdx.x;
  int wid  = tid >> 5;
  int lane = threadIdx.x & 31;
  int lo   = lane & 15;
  int hi   = lane >> 4;

  int chunk = wid & (NCHUNK - 1);
  int mt    = (wid >> 4) & (MTILES - 1);
  int l     = wid >> 10;
  int m0    = mt * 16;

  const float* zl = z + chunk * 65536 + l * 16384 + lo + hi * 2048;

  const float* eb = eS + (l * MCNT + m0 + lo) * DCNT + 2 * hi;
  v2f b0 = *(const v2f*)(eb + 0);
  v2f b1 = *(const v2f*)(eb + 4);
  v2f b2 = *(const v2f*)(eb + 8);
  v2f b3 = *(const v2f*)(eb + 12);

  float esq = esqS[l * MCNT + m0 + lo];
  v2f bS;
  bS.x = hi ? 0.0f : 1.0f;
  bS.y = hi ? 0.0f : esq;

  const float* zsql = zsqS + l * PCNT + chunk * HWCNT + lo;

  float p0 = 0.0f, p1 = 0.0f;
#pragma unroll 2
  for (int t = 0; t < PTILES; ++t) {
    const float* za = zl + 16 * t;
    v2f a0, a1, a2, a3;
    a0.x = za[0];          a0.y = za[1024];
    a1.x = za[4 * 1024];   a1.y = za[5 * 1024];
    a2.x = za[8 * 1024];   a2.y = za[9 * 1024];
    a3.x = za[12 * 1024];  a3.y = za[13 * 1024];

    float zq = zsql[16 * t];
    v2f aS;
    aS.x = hi ? 0.0f : zq;
    aS.y = hi ? 0.0f : 1.0f;

    v8f c = {0.f, 0.f, 0.f, 0.f, 0.f, 0.f, 0.f, 0.f};
    c = __builtin_amdgcn_wmma_f32_16x16x4_f32(false, a0, false, b0, (short)0, c, false, false);
    c = __builtin_amdgcn_wmma_f32_16x16x4_f32(false, a1, false, b1, (short)0, c, false, false);
    c = __builtin_amdgcn_wmma_f32_16x16x4_f32(false, a2, false, b2, (short)0, c, false, false);
    c = __builtin_amdgcn_wmma_f32_16x16x4_f32(false, a3, false, b3, (short)0, c, false, false);
    c = __builtin_amdgcn_wmma_f32_16x16x4_f32(false, aS, false, bS, (short)0, c, false, false);

    p0 += __builtin_amdgcn_exp2f(c[0]) + __builtin_amdgcn_exp2f(c[2])
        + __builtin_amdgcn_exp2f(c[4]) + __builtin_amdgcn_exp2f(c[6]);
    p1 += __builtin_amdgcn_exp2f(c[1]) + __builtin_amdgcn_exp2f(c[3])
        + __builtin_amdgcn_exp2f(c[5]) + __builtin_amdgcn_exp2f(c[7]);
  }
  float partial = p0 + p1;
  partial += __shfl_xor(partial, 16, 32);
  if (lane < 16)
    Spart[chunk * (LCNT * MCNT) + l * MCNT + m0 + lane] = partial;
}

// ---------------------------------------------------------------------------
// Finalize: S[l,m] = sum_chunk Spart; loss = -(1/(L*M)) * sum log(S)
//           + 0.5*z_dim*(2*log_sigma - 1) + log(P)
// ---------------------------------------------------------------------------
__global__ __launch_bounds__(256) void finalize_kernel(
    const float* __restrict__ Spart, const float* __restrict__ log_sigma,
    float* __restrict__ out)
{
  __shared__ float red[256];
  float s = 0.0f;
  for (int i = threadIdx.x; i < LCNT * MCNT; i += 256) {
    float acc = 0.0f;
#pragma unroll
    for (int c = 0; c < NCHUNK; ++c) acc += Spart[c * (LCNT * MCNT) + i];
    s += logf(acc);
  }
  red[threadIdx.x] = s;
  __syncthreads();
  for (int off = 128; off > 0; off >>= 1) {
    if (threadIdx.x < off) red[threadIdx.x] += red[threadIdx.x + off];
    __syncthreads();
  }
  if (threadIdx.x == 0) {
    float ls = log_sigma[0];
    out[0] = -(red[0] / (float)(LCNT * MCNT))
           + 0.5f * 64.0f * (2.0f * ls - 1.0f)
           + logf((float)PCNT);
  }
}

// ---------------------------------------------------------------------------
extern "C" void kernel_launch(void* const* d_in, const int* in_sizes, int n_in,
                              void* d_out, int out_size, void* d_ws, size_t ws_size,
                              hipStream_t stream) {
  (void)in_sizes; (void)n_in; (void)out_size;
  const float* z  = (const float*)d_in[0];   // 16*64*32*32
  const float* e  = (const float*)d_in[1];   // 4*1024*16
  const float* ls = (const float*)d_in[2];   // scalar
  float* out = (float*)d_out;

  float* ws    = (float*)d_ws;
  float* Spart = ws;                                  // 65536 f
  float* esqS  = ws + 65536;                          // 4096 f
  float* eS    = ws + 65536 + 4096;                   // 65536 f
  float* zsqS  = ws + 65536 + 4096 + 65536;           // 65536 f
  float* packedA = ws + 65536 + 4096 + 65536 + 65536; // 1310720 f

  const size_t need_packed =
      (size_t)(65536 + 4096 + 65536 + 65536 + 1310720) * sizeof(float); // ~6.0 MB

  prep_kernel<<<256, 256, 0, stream>>>(z, e, ls, esqS, eS, zsqS);
  if (ws_size >= need_packed) {
    pack_kernel<<<512, 256, 0, stream>>>(z, zsqS, packedA);
    vq_main_packed<<<512, 256, 0, stream>>>(packedA, eS, esqS, Spart);
  } else {
    vq_main_fallback<<<512, 256, 0, stream>>>(z, eS, esqS, zsqS, Spart);
  }
  finalize_kernel<<<1, 256, 0, stream>>>(Spart, ls, out);
}